// CenterLoss_90778428768344
// MI455X (gfx1250) — compile-verified
//
#include <hip/hip_runtime.h>
#include <math.h>

typedef __attribute__((ext_vector_type(2))) float v2f;
typedef __attribute__((ext_vector_type(4))) float v4f;
typedef __attribute__((ext_vector_type(8))) float v8f;

#define EPS_F 1e-6f
#define WAVES 8
#define BLOCK (WAVES * 32)
#define ROWS_PER_WAVE 16
#define ROWS_PER_BLOCK (WAVES * ROWS_PER_WAVE)

// Stage 1: each wave computes 16 row distances.
// Each lane accumulates squared diffs for row (lane&15) into two f32
// accumulators; (lane-half, acc#) maps onto the four K-slots of the
// V_WMMA_F32_16X16X4_F32 A-matrix. Since B is all-ones, D[m][n] = sum of the
// 4 slots of row m — one exact f32 WMMA does the whole 16-row reduction and
// broadcasts each row total across the C/D tile.
__global__ __launch_bounds__(BLOCK) void center_loss_partial(
    const float* __restrict__ x, const float* __restrict__ w,
    const int* __restrict__ t, float* __restrict__ partials, int N, int D) {
  __shared__ float s_wave[WAVES * 2];

  const int lane = threadIdx.x & 31;
  const int wave = threadIdx.x >> 5;
  const int rowInTile = lane & 15;  // A-matrix M index
  const int half = lane >> 4;       // selects K-slot pair {0,1} vs {2,3}
  const int rowBase = blockIdx.x * ROWS_PER_BLOCK + wave * ROWS_PER_WAVE;
  const int row = rowBase + rowInTile;
  const bool fullTile = (rowBase + ROWS_PER_WAVE <= N) && ((D & 7) == 0);

  float acc0 = 0.0f, acc1 = 0.0f;
  if (fullTile) {
    // Fast path: branch-free, b128 loads. Lane-half h covers columns
    // {c+4h .. c+4h+3} for c = 0,8,16,... — union covers all of D.
    const int cls = t[row];
    const float* __restrict__ xr = x + (size_t)row * (size_t)D + 4 * half;
    const float* __restrict__ wr = w + (size_t)cls * (size_t)D + 4 * half;
#pragma unroll 4
    for (int c = 0; c < D; c += 8) {
      v4f xv = __builtin_nontemporal_load((const v4f*)(xr + c));  // stream x
      v4f wv = *(const v4f*)(wr + c);                             // hot in L2
      float d0 = xv.x - wv.x;
      float d1 = xv.y - wv.y;
      float d2 = xv.z - wv.z;
      float d3 = xv.w - wv.w;
      acc0 = fmaf(d0, d0, acc0);
      acc1 = fmaf(d1, d1, acc1);
      acc0 = fmaf(d2, d2, acc0);
      acc1 = fmaf(d3, d3, acc1);
    }
  } else if (row < N) {
    // Generic tail path (never taken for N=65536, D=512).
    const int cls = t[row];
    const float* __restrict__ xr = x + (size_t)row * (size_t)D;
    const float* __restrict__ wr = w + (size_t)cls * (size_t)D;
    for (int c = 2 * half; c < D; c += 4) {
      float d0 = xr[c] - wr[c];
      acc0 = fmaf(d0, d0, acc0);
      if (c + 1 < D) {
        float d1 = xr[c + 1] - wr[c + 1];
        acc1 = fmaf(d1, d1, acc1);
      }
    }
  }

  // A = per-lane partials (16x4 f32), B = ones (layout-invariant), C = 0.
  v2f a; a.x = acc0; a.y = acc1;
  v2f b; b.x = 1.0f; b.y = 1.0f;
  v8f cacc = {};
  cacc = __builtin_amdgcn_wmma_f32_16x16x4_f32(
      /*neg_a=*/false, a, /*neg_b=*/false, b,
      /*c_mod=*/(short)0, cacc, /*reuse_a=*/false, /*reuse_b=*/false);

  // C/D layout: lane carries rows (v + 8*half) in VGPR v, replicated over N
  // columns; lanes 0 and 16 together hold all 16 row sums.
  float s = 0.0f;
  if (fullTile) {
#pragma unroll
    for (int v = 0; v < 8; ++v) s += __builtin_amdgcn_sqrtf(cacc[v] + EPS_F);
  } else {
#pragma unroll
    for (int v = 0; v < 8; ++v) {
      int grow = rowBase + v + 8 * half;
      if (grow < N) s += __builtin_amdgcn_sqrtf(cacc[v] + EPS_F);
    }
  }
  if ((lane & 15) == 0) s_wave[wave * 2 + half] = s;
  __syncthreads();

  if (threadIdx.x == 0) {
    float tot = 0.0f;
#pragma unroll
    for (int i = 0; i < WAVES * 2; ++i) tot += s_wave[i];
    partials[blockIdx.x] = tot;  // fixed-order, deterministic
  }
}

// Stage 2: deterministic sequential sum of block partials -> mean.
__global__ void center_loss_finalize(const float* __restrict__ partials,
                                     int nblocks, float* __restrict__ out,
                                     float invN) {
  if (threadIdx.x == 0 && blockIdx.x == 0) {
    float tot = 0.0f;
    for (int i = 0; i < nblocks; ++i) tot += partials[i];
    out[0] = tot * invN;
  }
}

extern "C" void kernel_launch(void* const* d_in, const int* in_sizes, int n_in,
                              void* d_out, int out_size, void* d_ws, size_t ws_size,
                              hipStream_t stream) {
  const float* x = (const float*)d_in[0];  // [N, D] f32
  const float* w = (const float*)d_in[1];  // [C, D] f32
  const int*   t = (const int*)d_in[2];    // [N] i32
  float* out = (float*)d_out;              // scalar f32

  const int N = in_sizes[2];
  const int D = in_sizes[0] / N;

  const int nblocks = (N + ROWS_PER_BLOCK - 1) / ROWS_PER_BLOCK;
  float* partials = (float*)d_ws;          // nblocks floats of scratch

  center_loss_partial<<<nblocks, BLOCK, 0, stream>>>(x, w, t, partials, N, D);
  center_loss_finalize<<<1, 32, 0, stream>>>(partials, nblocks, out,
                                             1.0f / (float)N);
}